// Clustering_82008105550130
// MI455X (gfx1250) — compile-verified
//
#include <hip/hip_runtime.h>
#include <hip/hip_bf16.h>
#include <math.h>

typedef __attribute__((ext_vector_type(2))) float v2f;
typedef __attribute__((ext_vector_type(8))) float v8f;

// ---------------------------------------------------------------------------
// Generic fp32 WMMA GEMM:  Out[b,m,n] = sum_k W[m,k] * X[b,k,n] + bias[m]
// One wave computes a 16(M) x 64(N) tile via 4x v_wmma_f32_16x16x4_f32 per
// k-step.  Block = 8 waves covering 16(M) x 512(N).  M,N,K are exact
// multiples of the tile sizes for every launch below (no bounds checks).
// Fragment layouts per CDNA5 ISA 7.12.2:
//   A 16x4 f32 : lane<16 -> {K0,K1}, lane>=16 -> {K2,K3}, row M = lane&15
//   B  4x16    : mirrored, col N = lane&15
//   C/D 16x16  : VGPR r -> M = r + 8*(lane>>4), N = lane&15
// ---------------------------------------------------------------------------
__global__ __launch_bounds__(256) void gemm_wmma_f32(
    const float* __restrict__ W, const float* __restrict__ bias,
    const float* __restrict__ X, float* __restrict__ Out,
    int M, int K, int N)
{
    const int lane   = threadIdx.x & 31;
    const int wave   = threadIdx.x >> 5;
    const int m_base = blockIdx.x * 16;
    const int n_base = blockIdx.y * 512 + wave * 64;
    const size_t xoff = (size_t)blockIdx.z * (size_t)K * (size_t)N;
    const size_t ooff = (size_t)blockIdx.z * (size_t)M * (size_t)N;
    const int mrow = lane & 15;
    const int ko   = (lane >> 4) << 1;              // 0 or 2

    const float* wr = W + (size_t)(m_base + mrow) * K + ko;
    const float* xb = X + xoff + (size_t)ko * N + (n_base + (lane & 15));

    v8f acc0 = {0.f,0.f,0.f,0.f,0.f,0.f,0.f,0.f};
    v8f acc1 = acc0, acc2 = acc0, acc3 = acc0;

    for (int k = 0; k < K; k += 4) {
        const float2 aw = *(const float2*)(wr + k);   // 8B aligned
        v2f a; a.x = aw.x; a.y = aw.y;
        const float* xk = xb + (size_t)k * N;
        v2f b0 = { xk[0],  xk[(size_t)N + 0]  };
        v2f b1 = { xk[16], xk[(size_t)N + 16] };
        v2f b2 = { xk[32], xk[(size_t)N + 32] };
        v2f b3 = { xk[48], xk[(size_t)N + 48] };
        acc0 = __builtin_amdgcn_wmma_f32_16x16x4_f32(false, a, false, b0, (short)0, acc0, false, false);
        acc1 = __builtin_amdgcn_wmma_f32_16x16x4_f32(false, a, false, b1, (short)0, acc1, false, false);
        acc2 = __builtin_amdgcn_wmma_f32_16x16x4_f32(false, a, false, b2, (short)0, acc2, false, false);
        acc3 = __builtin_amdgcn_wmma_f32_16x16x4_f32(false, a, false, b3, (short)0, acc3, false, false);
    }

    const int rext = (lane >> 4) * 8;
#pragma unroll
    for (int r = 0; r < 8; ++r) {
        const int row = m_base + r + rext;
        const float bv = bias[row];
        float* orow = Out + ooff + (size_t)row * N + n_base + (lane & 15);
        orow[0]  = acc0[r] + bv;
        orow[16] = acc1[r] + bv;
        orow[32] = acc2[r] + bv;
        orow[48] = acc3[r] + bv;
    }
}

// ---------------------------------------------------------------------------
// Adaptive 2x2 pool over (64,64) spatial per (n,c):
//   cp[n,c,i,j] = mean_{32x32 block} xq[n,c,...],  same for feature -> cfp.
// quad row bit == (e>=8), quad col bit == (t>>5)&1 (derived statically).
// Deterministic serial reduction by threads 0..3.
// ---------------------------------------------------------------------------
__global__ __launch_bounds__(256) void pool_kernel(
    const float* __restrict__ xq, const float* __restrict__ fea,
    float* __restrict__ cp, float* __restrict__ cfp)
{
    const int c = blockIdx.x;        // 0..63
    const int n = blockIdx.y;        // 0..255
    const int t = threadIdx.x;
    const size_t base = (size_t)n * 262144 + (size_t)c * 4096;

    float q0 = 0.f, q1 = 0.f, f0 = 0.f, f1 = 0.f;
#pragma unroll
    for (int e = 0; e < 8; ++e)  { int sp = e*256 + t; q0 += xq[base+sp]; f0 += fea[base+sp]; }
#pragma unroll
    for (int e = 8; e < 16; ++e) { int sp = e*256 + t; q1 += xq[base+sp]; f1 += fea[base+sp]; }

    __shared__ float rq0[256], rq1[256], rf0[256], rf1[256];
    rq0[t] = q0; rq1[t] = q1; rf0[t] = f0; rf1[t] = f1;
    __syncthreads();

    if (t < 4) {
        const int col = t & 1, rowsel = t >> 1;
        float sq = 0.f, sf = 0.f;
        for (int i = 0; i < 256; ++i) {
            if (((i >> 5) & 1) == col) {
                sq += rowsel ? rq1[i] : rq0[i];
                sf += rowsel ? rf1[i] : rf0[i];
            }
        }
        const size_t o = (size_t)n * 256 + (size_t)c * 4 + t; // quad = i*2+j = t
        cp[o]  = sq * (1.f / 1024.f);
        cfp[o] = sf * (1.f / 1024.f);
    }
}

// ---------------------------------------------------------------------------
// Cross attention, one block per (u,hh) problem (256 blocks):
//   q  = cp   viewed (32,4,8,64)     [4 queries, pre-scaled by 1/8]
//   K  = val  viewed (32,4096,8,64)
//   V  = fea  viewed (32,4096,8,64)
//   out -> ch viewed (32,4,8,64)
// Scores staged in global ws (4x4096 floats/block). Deterministic tree
// reductions for row max / row sum.
// ---------------------------------------------------------------------------
__global__ __launch_bounds__(256) void attn_kernel(
    const float* __restrict__ cp, const float* __restrict__ val,
    const float* __restrict__ fea, float* __restrict__ ch,
    float* __restrict__ S)
{
    const int bid = blockIdx.x;            // 0..255
    const int u = bid >> 3, hh = bid & 7;
    const int t = threadIdx.x;

    __shared__ float qs[4][64];
    __shared__ float rbuf[4][256];
    __shared__ float mxs[4], sms[4];
    __shared__ __align__(16) float ored[4][4][64];

    { int m = t >> 6, c = t & 63;
      qs[m][c] = cp[(size_t)u*2048 + (size_t)m*512 + (size_t)hh*64 + c] * 0.125f; }
    __syncthreads();

    float* Sb = S + (size_t)bid * 4 * 4096;
    const float* kb = val + (size_t)u * 2097152 + (size_t)hh * 64;

    float lmax[4] = {-1e30f,-1e30f,-1e30f,-1e30f};
    for (int e = 0; e < 16; ++e) {
        const int n = e*256 + t;
        const float4* kr = (const float4*)(kb + (size_t)n * 512);
        float s0 = 0.f, s1 = 0.f, s2 = 0.f, s3 = 0.f;
#pragma unroll
        for (int c4 = 0; c4 < 16; ++c4) {
            const float4 kv = kr[c4];
            s0 += qs[0][c4*4]*kv.x + qs[0][c4*4+1]*kv.y + qs[0][c4*4+2]*kv.z + qs[0][c4*4+3]*kv.w;
            s1 += qs[1][c4*4]*kv.x + qs[1][c4*4+1]*kv.y + qs[1][c4*4+2]*kv.z + qs[1][c4*4+3]*kv.w;
            s2 += qs[2][c4*4]*kv.x + qs[2][c4*4+1]*kv.y + qs[2][c4*4+2]*kv.z + qs[2][c4*4+3]*kv.w;
            s3 += qs[3][c4*4]*kv.x + qs[3][c4*4+1]*kv.y + qs[3][c4*4+2]*kv.z + qs[3][c4*4+3]*kv.w;
        }
        Sb[0*4096+n] = s0; Sb[1*4096+n] = s1; Sb[2*4096+n] = s2; Sb[3*4096+n] = s3;
        lmax[0] = fmaxf(lmax[0], s0); lmax[1] = fmaxf(lmax[1], s1);
        lmax[2] = fmaxf(lmax[2], s2); lmax[3] = fmaxf(lmax[3], s3);
    }

#pragma unroll
    for (int m = 0; m < 4; ++m) rbuf[m][t] = lmax[m];
    __syncthreads();
    for (int st = 128; st > 0; st >>= 1) {
        if (t < st) {
#pragma unroll
            for (int m = 0; m < 4; ++m) rbuf[m][t] = fmaxf(rbuf[m][t], rbuf[m][t+st]);
        }
        __syncthreads();
    }
    if (t < 4) mxs[t] = rbuf[t][0];
    __syncthreads();

    float lsum[4] = {0.f,0.f,0.f,0.f};
    for (int e = 0; e < 16; ++e) {
        const int n = e*256 + t;
#pragma unroll
        for (int m = 0; m < 4; ++m) {
            const float p = __expf(Sb[m*4096+n] - mxs[m]);
            Sb[m*4096+n] = p; lsum[m] += p;
        }
    }
#pragma unroll
    for (int m = 0; m < 4; ++m) rbuf[m][t] = lsum[m];
    __syncthreads();
    for (int st = 128; st > 0; st >>= 1) {
        if (t < st) {
#pragma unroll
            for (int m = 0; m < 4; ++m) rbuf[m][t] += rbuf[m][t+st];
        }
        __syncthreads();
    }
    if (t < 4) sms[t] = rbuf[t][0];
    __syncthreads();

    const int grp = t >> 6, c = t & 63;
    const float* vcol = fea + (size_t)u * 2097152 + (size_t)hh * 64 + c;
    float a0 = 0.f, a1 = 0.f, a2 = 0.f, a3 = 0.f;
    for (int n = grp*1024; n < grp*1024 + 1024; ++n) {
        const float v = vcol[(size_t)n * 512];
        a0 += Sb[0*4096+n]*v; a1 += Sb[1*4096+n]*v;
        a2 += Sb[2*4096+n]*v; a3 += Sb[3*4096+n]*v;
    }
    ored[grp][0][c] = a0; ored[grp][1][c] = a1;
    ored[grp][2][c] = a2; ored[grp][3][c] = a3;
    __syncthreads();
    if (grp == 0) {
#pragma unroll
        for (int m = 0; m < 4; ++m) {
            const float o = (ored[0][m][c] + ored[1][m][c] + ored[2][m][c] + ored[3][m][c]) / sms[m];
            ch[(size_t)u*2048 + (size_t)m*512 + (size_t)hh*64 + c] = o;
        }
    }
}

// ---------------------------------------------------------------------------
// Clustering stage, one block per g (64 blocks):
//  qhat[k][c] = normalize_c( ch[nq(k)*256 + c*4 + i*2 + j] )      (16x64)
//  cf[k][c]   = cfp[same index]                                   (16x64)
//  per n in [0,4096): p = xq[n_g, c, n>>6, n&63]; cosine sim vs each k,
//    sigmoid(beta + alpha*sim); first-argmax -> (sval[n], idxs[n]), cnt.
//  agg[k][c]  = sum_{idx==k} sval[n]*tgf(g,n,c)  (deterministic k,c-major)
//  out_ = (agg + cf)/(cnt+1);  Y[(g*4096+n)*64+c] = sval[n]*out_[idx[n]][c]
//  tgf(g,n,c) = fea[ Xg*2097152 + (n&63)*32768 + c*512 + hh*64 + (n>>6) ]
// ---------------------------------------------------------------------------
__global__ __launch_bounds__(256) void cluster_kernel(
    const float* __restrict__ ch, const float* __restrict__ cfp,
    const float* __restrict__ xq, const float* __restrict__ fea,
    const float* __restrict__ alpha_p, const float* __restrict__ beta_p,
    float* __restrict__ Y)
{
    const int g = blockIdx.x;                  // 0..63
    const int t = threadIdx.x;
    const int a = g >> 5, hh = (g >> 2) & 7, w4 = g & 3;

    __shared__ __align__(16) float qhat[16][64];
    __shared__ __align__(16) float cf[16][64];
    __shared__ __align__(16) float agg[16][64];
    __shared__ float sval[4096];
    __shared__ unsigned char idxs[4096];
    __shared__ int cnt[16];

    if (t < 16) cnt[t] = 0;
    if (t < 16) {
        const int k = t, tt = k >> 2, ii = (k >> 1) & 1, jj = k & 1;
        const size_t nq = (size_t)(a*128 + tt*32 + hh*4 + w4);
        const size_t base = nq*256 + (size_t)(ii*2 + jj);
        float ss = 0.f;
        for (int c = 0; c < 64; ++c) { const float v = ch[base + (size_t)c*4]; qhat[k][c] = v; ss += v*v; }
        const float inv = 1.f / fmaxf(sqrtf(ss), 1e-12f);
        for (int c = 0; c < 64; ++c) qhat[k][c] *= inv;
        for (int c = 0; c < 64; ++c) cf[k][c] = cfp[base + (size_t)c*4];
    }
    __syncthreads();

    const float alpha = alpha_p[0], beta = beta_p[0];
    const size_t ng = (size_t)(a*128 + 96 + hh*4 + w4);       // tt = t-1 = 3
    const float* xqb = xq + ng * 262144;
    const int Xg = a*16 + 12 + w4;
    const float* fbase = fea + (size_t)Xg * 2097152 + (size_t)hh * 64;

    for (int e = 0; e < 16; ++e) {
        const int n = e*256 + t;
        float pr[64]; float ss = 0.f;
#pragma unroll
        for (int c = 0; c < 64; ++c) { const float v = xqb[(size_t)c*4096 + n]; pr[c] = v; ss += v*v; }
        const float inv = 1.f / fmaxf(sqrtf(ss), 1e-12f);
        float best = -1e30f; int bi = 0;
        for (int k = 0; k < 16; ++k) {
            float d = 0.f;
#pragma unroll
            for (int c = 0; c < 64; ++c) d += qhat[k][c] * pr[c];
            const float sv = 1.f / (1.f + __expf(-(beta + alpha * d * inv)));
            if (sv > best) { best = sv; bi = k; }   // first max (numpy argmax)
        }
        sval[n] = best; idxs[n] = (unsigned char)bi;
        atomicAdd(&cnt[bi], 1);                      // integer: deterministic
    }
    __syncthreads();

    // deterministic agg + finalize: out_[k][c] = (agg+cf)/(cnt+1)
    for (int i = t; i < 1024; i += 256) {
        const int k = i >> 6, c = i & 63;
        float acc = 0.f;
        for (int n = 0; n < 4096; ++n) {
            if (idxs[n] == (unsigned char)k)
                acc += sval[n] * fbase[(size_t)(n & 63)*32768 + (size_t)c*512 + (n >> 6)];
        }
        agg[k][c] = (acc + cf[k][c]) / ((float)cnt[k] + 1.f);
    }
    __syncthreads();

    for (int e = 0; e < 16; ++e) {
        const int n = e*256 + t;
        const float s = sval[n];
        const int k = idxs[n];
        float4* yo = (float4*)(Y + ((size_t)g*4096 + n) * 64);
        const float4* ar = (const float4*)agg[k];
#pragma unroll
        for (int c4 = 0; c4 < 16; ++c4) {
            const float4 v = ar[c4];
            yo[c4] = make_float4(v.x*s, v.y*s, v.z*s, v.w*s);
        }
    }
}

// ---------------------------------------------------------------------------
// Workspace layout (floats):
//   xq   67,108,864 | val 67,108,864 | fea 67,108,864
//   cp 65,536 | cfp 65,536 | ch 65,536 | S 4,194,304 | Y 16,777,216
// total ~890 MB.
// ---------------------------------------------------------------------------
extern "C" void kernel_launch(void* const* d_in, const int* in_sizes, int n_in,
                              void* d_out, int out_size, void* d_ws, size_t ws_size,
                              hipStream_t stream) {
    const float* x  = (const float*)d_in[0];
    const float* w1 = (const float*)d_in[1]; const float* b1 = (const float*)d_in[2];
    const float* wv = (const float*)d_in[3]; const float* bv = (const float*)d_in[4];
    const float* wf = (const float*)d_in[5]; const float* bf = (const float*)d_in[6];
    const float* w2 = (const float*)d_in[7]; const float* b2 = (const float*)d_in[8];
    const float* al = (const float*)d_in[9]; const float* be = (const float*)d_in[10];

    float* ws  = (float*)d_ws;
    float* xq  = ws;
    float* val = xq  + 67108864ull;
    float* fea = val + 67108864ull;
    float* cp  = fea + 67108864ull;
    float* cfp = cp  + 65536ull;
    float* chb = cfp + 65536ull;
    float* S   = chb + 65536ull;
    float* Y   = S   + 4194304ull;
    float* out = (float*)d_out;

    const dim3 blk(256);
    // 3 fused-1x1-conv GEMMs: M=512, K=256, N=16384, batch=8
    gemm_wmma_f32<<<dim3(32, 32, 8), blk, 0, stream>>>(w1, b1, x, xq,  512, 256, 16384);
    gemm_wmma_f32<<<dim3(32, 32, 8), blk, 0, stream>>>(wv, bv, x, val, 512, 256, 16384);
    gemm_wmma_f32<<<dim3(32, 32, 8), blk, 0, stream>>>(wf, bf, x, fea, 512, 256, 16384);
    // adaptive pool (xq -> cp, fea -> cfp)
    pool_kernel<<<dim3(64, 256), blk, 0, stream>>>(xq, fea, cp, cfp);
    // clustering attention: centers_h = attn(cp, val, fea)
    attn_kernel<<<dim3(256), blk, 0, stream>>>(cp, val, fea, chb, S);
    // cosine-sim assignment + aggregation -> Y (pre-conv2 activations)
    cluster_kernel<<<dim3(64), blk, 0, stream>>>(chb, cfp, xq, fea, al, be, Y);
    // conv2: M=256, K=512, N=16384, batch=2 -> d_out (2,256,128,128)
    gemm_wmma_f32<<<dim3(16, 32, 2), blk, 0, stream>>>(w2, b2, Y, out, 256, 512, 16384);
}